// SplineFeaturePropagation_13778255085921
// MI455X (gfx1250) — compile-verified
//
#include <hip/hip_runtime.h>
#include <stdint.h>

// ---------------------------------------------------------------------------
// SplineFeaturePropagation: out[row[e]] += exp(-attr[e]) * x[col[e]]
// N_NODES=100000, N_EDGES=1600000, D=64.  Memory/atomic bound (AI ~0.25 F/B);
// x (25.6MB) and out (25.6MB) are L2-resident on MI455X (192MB L2).
//
// Strategy (gfx1250, wave32):
//   * 16 lanes per edge, one float4 (b128) chunk per lane, 2 edges per wave.
//   * Edge metadata (row,col,attr) staged per-wave into LDS with
//     global_load_async_to_lds_b32, TRIPLE-buffered, synced by s_wait_asynccnt.
//     Pipeline: wait(k,k+1 ready) -> issue k+2 -> prefetch k+1 gather rows
//     (global_prefetch_b8, one full batch of lead time) -> process k.
//   * Per-edge weight exp(-attr) computed once per edge (not per lane-use),
//     wave-synchronously cached in LDS.
//   * Scatter via global_atomic_add_f32 (f32 has no packed atomic; 4/lane is
//     minimal). ~16 edges/node average => negligible contention.
// ---------------------------------------------------------------------------

#define D_FEAT        64
#define WAVE          32
#define EDGE_BATCH    32             // edges staged per wave per pipeline stage
#define NBUF          3              // triple-buffered metadata pipeline
#define WPB           8              // waves per block
#define BLOCK_THREADS (WAVE * WPB)   // 256

__device__ __forceinline__ void async_load_b32(uint32_t lds_addr, uint32_t goff,
                                               const void* sbase) {
  // GVS form: mem = SGPR64 + VGPR32 ; VDST holds LDS byte address.
  asm volatile("global_load_async_to_lds_b32 %0, %1, %2"
               :
               : "v"(lds_addr), "v"(goff), "s"(sbase)
               : "memory");
}

__device__ __forceinline__ void wait_async0() {
  asm volatile("s_wait_asynccnt 0" ::: "memory");
}

__device__ __forceinline__ void wave_lds_fence() {
  // All 32 lanes of a wave issue DS ops together and DS is in-order per wave;
  // only the compiler must not reorder around cross-lane LDS communication.
  asm volatile("" ::: "memory");
}

__global__ void zero_out_kernel(float4* __restrict__ out, int n4) {
  int i = blockIdx.x * blockDim.x + threadIdx.x;
  if (i < n4) out[i] = make_float4(0.f, 0.f, 0.f, 0.f);
}

__global__ __launch_bounds__(BLOCK_THREADS)
void spline_scatter_kernel(const float* __restrict__ x,
                           const int*   __restrict__ ei,   // [2,E]: rows, cols
                           const float* __restrict__ ea,   // [E]
                           float*       __restrict__ out,
                           int E)
{
  // Per-wave triple-buffered metadata: [wave][buf][{row,col,attr->w}][edge]
  __shared__ uint32_t smeta[WPB][NBUF][3][EDGE_BATCH];

  const int lane   = threadIdx.x & (WAVE - 1);
  const int wav    = threadIdx.x >> 5;
  const int gwave  = blockIdx.x * WPB + wav;
  const int nwaves = gridDim.x * WPB;
  const int nbatch = (E + EDGE_BATCH - 1) / EDGE_BATCH;

  const int* rowp = ei;        // edge_index[0][:]
  const int* colp = ei + E;    // edge_index[1][:]

  // Low 32 bits of a flat pointer to __shared__ == LDS byte offset.
  uint32_t ldsb[NBUF];
#pragma unroll
  for (int b = 0; b < NBUF; ++b)
    ldsb[b] = (uint32_t)(uintptr_t)(&smeta[wav][b][0][0]) + (uint32_t)lane * 4u;

  const int sub  = lane & 15;  // which float4 chunk of the 64-wide row
  const int half = lane >> 4;  // which edge of the pair this iteration

  // Stage a batch of 32 edges' metadata into LDS buffer `b` (async, per-lane).
  auto issue = [&](int batch, int b) {
    int e = batch * EDGE_BATCH + lane;
    if (e >= E) e = E - 1;                       // clamp; masked at compute
    uint32_t go = (uint32_t)e * 4u;
    async_load_b32(ldsb[b],                        go, rowp);
    async_load_b32(ldsb[b] + EDGE_BATCH * 4u,      go, colp);
    async_load_b32(ldsb[b] + 2u * EDGE_BATCH * 4u, go, ea);
  };

  int k = gwave;
  if (k          < nbatch) issue(k,          0);
  if (k + nwaves < nbatch) issue(k + nwaves, 1);

  int bi = 0;
  for (; k < nbatch; k += nwaves) {
    wait_async0();                    // batches k and k+1 resident in LDS

    int bi1 = bi + 1;  if (bi1 == NBUF) bi1 = 0;
    int bi2 = bi1 + 1; if (bi2 == NBUF) bi2 = 0;

    // Keep the async stage two batches deep.
    int k2 = k + 2 * nwaves;
    if (k2 < nbatch) issue(k2, bi2);

    // Prefetch NEXT batch's gather rows (256B row = 2 cache lines): a full
    // batch of gather+atomic work elapses before they are consumed.
    if (k + nwaves < nbatch) {
      int e = (k + nwaves) * EDGE_BATCH + lane;
      if (e < E) {
        int c = (int)smeta[wav][bi1][1][lane];
        const char* p = (const char*)(x + (size_t)c * D_FEAT);
        __builtin_prefetch(p, 0, 0);
        __builtin_prefetch(p + 128, 0, 0);
      }
    }

    // Per-edge weight, computed exactly once (overwrite attr slot with w).
    {
      int e = k * EDGE_BATCH + lane;
      if (e < E) {
        float a = __uint_as_float(smeta[wav][bi][2][lane]);
        smeta[wav][bi][2][lane] = __float_as_uint(__expf(-a));
      }
    }
    wave_lds_fence();

    const int base = k * EDGE_BATCH;
#pragma unroll 4
    for (int it = 0; it < EDGE_BATCH / 2; ++it) {
      const int slot = 2 * it + half;  // half-wave 0 -> even edge, 1 -> odd
      const int e = base + slot;
      if (e < E) {
        // Same-address LDS reads broadcast across a half-wave's 16 lanes.
        const int   r = (int)smeta[wav][bi][0][slot];
        const int   c = (int)smeta[wav][bi][1][slot];
        const float w = __uint_as_float(smeta[wav][bi][2][slot]);

        const float4 v = *(const float4*)(x + (size_t)c * D_FEAT + sub * 4);
        float* o = out + (size_t)r * D_FEAT + sub * 4;
        atomicAdd(o + 0, w * v.x);
        atomicAdd(o + 1, w * v.y);
        atomicAdd(o + 2, w * v.z);
        atomicAdd(o + 3, w * v.w);
      }
    }
    bi = bi1;
  }
}

extern "C" void kernel_launch(void* const* d_in, const int* in_sizes, int n_in,
                              void* d_out, int out_size, void* d_ws, size_t ws_size,
                              hipStream_t stream) {
  const float* x   = (const float*)d_in[0];  // [N_NODES, 64] f32
  const int*   ei  = (const int*)d_in[1];    // [2, E] int
  const float* ea  = (const float*)d_in[2];  // [E] f32
  float*       out = (float*)d_out;          // [N_NODES, 64] f32
  (void)n_in; (void)d_ws; (void)ws_size;

  const int E  = in_sizes[2];
  const int n4 = out_size / 4;

  // 1) zero-fill the (poisoned) output, vectorized b128 stores.
  zero_out_kernel<<<(n4 + 255) / 256, 256, 0, stream>>>((float4*)out, n4);

  // 2) edge scatter-add with triple-buffered async-LDS metadata pipeline.
  const int nbatch = (E + EDGE_BATCH - 1) / EDGE_BATCH;
  int blocks = (nbatch + WPB - 1) / WPB;
  if (blocks > 768) blocks = 768;            // ~6K waves: deep pipeline, full chip
  if (blocks < 1)   blocks = 1;
  spline_scatter_kernel<<<blocks, BLOCK_THREADS, 0, stream>>>(x, ei, ea, out, E);
}